// DualPathBackbone_32323923869828
// MI455X (gfx1250) — compile-verified
//
#include <hip/hip_runtime.h>
#include <hip/hip_bf16.h>

// Problem constants from the reference: B,H,W = 4,64,2048 ; C = 128.
constexpr int Bc = 4;
constexpr int Hc = 64;
constexpr int Wc = 2048;
constexpr int Cc = 128;                         // features per point
constexpr int NUM_CELLS = Bc * Hc * Wc;         // 524288 grid cells
constexpr int WAVE = 32;                        // gfx1250 is wave32-only

typedef __attribute__((ext_vector_type(4))) float v4f;

// Hardware no-return f32 atomic add (STOREcnt path, no CAS loop, no VGPR
// writeback). Device scope so the following kernel observes the sums.
__device__ __forceinline__ void atomic_add_f32_hw(float* addr, float v) {
    asm volatile("global_atomic_add_f32 %0, %1, off scope:SCOPE_DEV"
                 :: "v"(addr), "v"(v)
                 : "memory");
}

// ---------------------------------------------------------------------------
// Kernel 1: zero the workspace (sums [NUM_CELLS*C] + counts [NUM_CELLS]).
// Regular-temporal b128 stores: we WANT these lines resident in the 192MB L2
// because the next kernel does 128M f32 atomic RMWs against them (atomics
// execute at L2).
// ---------------------------------------------------------------------------
__global__ __launch_bounds__(256)
void zero_ws_kernel(v4f* __restrict__ ws, long n_vec4) {
    long i = (long)blockIdx.x * blockDim.x + threadIdx.x;
    long stride = (long)gridDim.x * blockDim.x;
    v4f z = (v4f)(0.0f);
    for (; i < n_vec4; i += stride) {
        ws[i] = z;                                // RT store: keep in L2
    }
}

// ---------------------------------------------------------------------------
// Kernel 2: scatter. One wave32 per point; lane L owns features [4L, 4L+4).
//  - point_features row: one NT global_load_b128 per lane (read-once stream,
//    keep the 256MB sums array resident in L2 instead)
//  - 4x global_atomic_add_f32 (no-return) per lane into sums
//  - lane 0 bumps counts[cell]
//  - global_prefetch_b8 of the next grid-stride iteration's feature row
// ---------------------------------------------------------------------------
__global__ __launch_bounds__(256)
void scatter_kernel(const float* __restrict__ pf,
                    const int* __restrict__ cb,
                    const int* __restrict__ cy,
                    const int* __restrict__ cx,
                    float* __restrict__ sums,
                    float* __restrict__ counts,
                    int N) {
    const int lane     = threadIdx.x & (WAVE - 1);
    const int waveId   = (int)((blockIdx.x * blockDim.x + threadIdx.x) >> 5);
    const int numWaves = (int)((gridDim.x * blockDim.x) >> 5);

    for (int n = waveId; n < N; n += numWaves) {
        // Prefetch next iteration's 512B feature row into cache early.
        int n2 = n + numWaves;
        if (n2 < N) {
            __builtin_prefetch(pf + (size_t)n2 * Cc + lane * 4, 0, 0); // -> global_prefetch_b8
        }

        const int b = cb[n], y = cy[n], x = cx[n];
        const int cell = (b * Hc + y) * Wc + x;

        // Non-temporal read-once load of this lane's 4 features.
        v4f v = __builtin_nontemporal_load(
            (const v4f*)(pf + (size_t)n * Cc + lane * 4));

        float* dst = sums + (size_t)cell * Cc + lane * 4;
        atomic_add_f32_hw(dst + 0, v.x);
        atomic_add_f32_hw(dst + 1, v.y);
        atomic_add_f32_hw(dst + 2, v.z);
        atomic_add_f32_hw(dst + 3, v.w);

        if (lane == 0) {
            atomic_add_f32_hw(counts + cell, 1.0f);
        }
    }
}

// ---------------------------------------------------------------------------
// Kernel 3: fused mean + gather. One wave32 per point.
//  - counts[cell] load broadcasts across the wave (same address all lanes)
//  - sums rows are hot in L2 (they were just atomically built there)
//  - output is written once with NT stores so the 512MB stream doesn't
//    evict sums rows other waves still need
// ---------------------------------------------------------------------------
__global__ __launch_bounds__(256)
void gather_kernel(const float* __restrict__ sums,
                   const float* __restrict__ counts,
                   const int* __restrict__ cb,
                   const int* __restrict__ cy,
                   const int* __restrict__ cx,
                   float* __restrict__ out,
                   int N) {
    const int lane     = threadIdx.x & (WAVE - 1);
    const int waveId   = (int)((blockIdx.x * blockDim.x + threadIdx.x) >> 5);
    const int numWaves = (int)((gridDim.x * blockDim.x) >> 5);

    for (int n = waveId; n < N; n += numWaves) {
        const int b = cb[n], y = cy[n], x = cx[n];
        const int cell = (b * Hc + y) * Wc + x;

        const float cnt = counts[cell];
        const float inv = 1.0f / fmaxf(cnt, 1.0f);

        v4f s = *(const v4f*)(sums + (size_t)cell * Cc + lane * 4);
        v4f r;
        r.x = s.x * inv;
        r.y = s.y * inv;
        r.z = s.z * inv;
        r.w = s.w * inv;

        __builtin_nontemporal_store(r, (v4f*)(out + (size_t)n * Cc + lane * 4));
    }
}

// ---------------------------------------------------------------------------
// Launch
// d_in order: point_features [N*C] f32, coors_b [N] i32, coors_y [N] i32,
//             coors_x [N] i32.  d_out: [N*C] f32.
// d_ws layout: sums f32[NUM_CELLS*C] | counts f32[NUM_CELLS]  (~270 MB)
// ---------------------------------------------------------------------------
extern "C" void kernel_launch(void* const* d_in, const int* in_sizes, int n_in,
                              void* d_out, int out_size, void* d_ws, size_t ws_size,
                              hipStream_t stream) {
    const float* pf = (const float*)d_in[0];
    const int*   cb = (const int*)d_in[1];
    const int*   cy = (const int*)d_in[2];
    const int*   cx = (const int*)d_in[3];
    float*       out = (float*)d_out;

    const int N = in_sizes[1];                       // number of points

    float* sums   = (float*)d_ws;
    float* counts = sums + (size_t)NUM_CELLS * Cc;

    // 1) zero sums + counts (RT stores -> resident in L2 for the atomic phase)
    const long n_floats = (long)NUM_CELLS * (Cc + 1); // divisible by 4
    const long n_vec4   = n_floats / 4;
    zero_ws_kernel<<<4096, 256, 0, stream>>>((v4f*)d_ws, n_vec4);

    // 2) scatter-add (one wave per point, grid-stride)
    // 2048 blocks * 256 thr = 16384 waves; ~61 points per wave at N=1M.
    scatter_kernel<<<2048, 256, 0, stream>>>(pf, cb, cy, cx, sums, counts, N);

    // 3) fused mean + gather
    gather_kernel<<<2048, 256, 0, stream>>>(sums, counts, cb, cy, cx, out, N);
}